// RNN_72490458022211
// MI455X (gfx1250) — compile-verified
//
#include <hip/hip_runtime.h>
#include <hip/hip_bf16.h>

typedef __attribute__((ext_vector_type(2))) float v2f;
typedef __attribute__((ext_vector_type(4))) float v4f;
typedef __attribute__((ext_vector_type(8))) float v8f;

__device__ __forceinline__ float fast_tanh(float v) {
#if __has_builtin(__builtin_amdgcn_tanhf)
  return __builtin_amdgcn_tanhf(v);          // gfx1250 v_tanh_f32
#elif __has_builtin(__builtin_amdgcn_tanh_f32)
  return __builtin_amdgcn_tanh_f32(v);
#else
  return tanhf(v);
#endif
}

// One wave (32 lanes) owns a 16-batch-row tile and runs the full T-step
// recurrence. Per step: 5 chained V_WMMA_F32_16X16X4_F32 (D->C accumulation
// is the pipelined path; no D->A/B hazard in the chain):
//   acc  = [x_t | 1 | 0 | 0] x [w_ih ; b_ih+b_hh ; 0 ; 0]   (input proj + bias)
//   acc += H[:,4c:4c+4] x W_hh^T[4c:4c+4,:]  for c = 0..3    (recurrent matmul)
// then h = tanh(acc), transposed D->A layout via a 1KB per-wave LDS tile
// (same-wave LDS ops are processed in order -> no barriers needed).
// x is double-buffered one 4-step block ahead so the global-load latency is
// off the serial critical path.
__global__ __launch_bounds__(32) void rnn_wmma_f32_kernel(
    const float* __restrict__ x,     // [B, T]
    const float* __restrict__ w_ih,  // [16]
    const float* __restrict__ w_hh,  // [16,16] row-major w_hh[j][k]
    const float* __restrict__ b_ih,  // [16]
    const float* __restrict__ b_hh,  // [16]
    const float* __restrict__ w_fc,  // [16]
    const float* __restrict__ b_fc,  // [1]
    float* __restrict__ out,         // [B]
    int T) {
  __shared__ float hs[16 * 16];  // hs[m][j]: hidden state, m = batch row in tile

  const int lane = threadIdx.x & 31;
  const int ln   = lane & 15;   // A-operand row (batch row) / B,C,D column
  const int half = lane >> 4;
  const int row  = blockIdx.x * 16 + ln;  // global batch row

  // --- constant B operands (resident in VGPRs for the whole recurrence) ---
  // B_c[vgpr] holds W_hh^T[4c + 2*half + vgpr][ln] = w_hh[ln][4c + 2*half + vgpr]
  v2f Bw[4];
#pragma unroll
  for (int c = 0; c < 4; ++c)
    Bw[c] = *(const v2f*)(w_hh + ln * 16 + c * 4 + half * 2);
  // Input-projection B: rows [w_ih ; b_ih+b_hh ; 0 ; 0]
  v2f Bx;
  Bx.x = half ? 0.f : w_ih[ln];
  Bx.y = half ? 0.f : (b_ih[ln] + b_hh[ln]);

  // zero initial hidden state
  for (int i = lane; i < 256; i += 32) hs[i] = 0.f;
  __builtin_amdgcn_wave_barrier();

  const float* xrow = x + (size_t)row * T;

  // software pipeline: xq always holds the block for the *current* t4
  v4f xq = *(const v4f*)(xrow);

  for (int t4 = 0; t4 < T; t4 += 4) {
    const v4f xc = xq;
    // prefetch next block (last iteration harmlessly reloads the row start)
    const int tn = (t4 + 4 < T) ? (t4 + 4) : 0;
    xq = *(const v4f*)(xrow + tn);

#pragma unroll
    for (int u = 0; u < 4; ++u) {
      // A for input projection: row ln = [x_t, 1, 0, 0]
      v2f A4;
      A4.x = half ? 0.f : xc[u];
      A4.y = half ? 0.f : 1.0f;
      v8f acc = {};
      acc = __builtin_amdgcn_wmma_f32_16x16x4_f32(false, A4, false, Bx,
                                                  (short)0, acc, false, false);
#pragma unroll
      for (int c = 0; c < 4; ++c) {
        // A chunk: H[ln][4c + 2*half .. +1]  (8-byte LDS load, in-order vs stores)
        v2f Ac = *(const v2f*)(hs + ln * 16 + c * 4 + half * 2);
        acc = __builtin_amdgcn_wmma_f32_16x16x4_f32(false, Ac, false, Bw[c],
                                                    (short)0, acc, false, false);
      }
      __builtin_amdgcn_wave_barrier();  // reads of step t before overwrite
      // D layout: VGPR v, lane -> (m = v + 8*half, j = ln); apply tanh, store
#pragma unroll
      for (int v = 0; v < 8; ++v)
        hs[(v + half * 8) * 16 + ln] = fast_tanh(acc[v]);
      __builtin_amdgcn_wave_barrier();  // stores before step t+1 reads
    }
  }

  // final linear: out[row] = h_last[row,:] . w_fc + b_fc (lanes 0..15 only)
  if (half == 0) {
    float s = b_fc[0];
#pragma unroll
    for (int j = 0; j < 16; j += 4) {
      v4f hq = *(const v4f*)(hs + ln * 16 + j);
      v4f wq = *(const v4f*)(w_fc + j);
      s += hq[0] * wq[0] + hq[1] * wq[1] + hq[2] * wq[2] + hq[3] * wq[3];
    }
    out[row] = s;
  }
}

extern "C" void kernel_launch(void* const* d_in, const int* in_sizes, int n_in,
                              void* d_out, int out_size, void* d_ws, size_t ws_size,
                              hipStream_t stream) {
  (void)n_in; (void)d_ws; (void)ws_size;
  const float* x    = (const float*)d_in[0];
  const float* w_ih = (const float*)d_in[1];
  const float* w_hh = (const float*)d_in[2];
  const float* b_ih = (const float*)d_in[3];
  const float* b_hh = (const float*)d_in[4];
  const float* w_fc = (const float*)d_in[5];
  const float* b_fc = (const float*)d_in[6];
  float* out = (float*)d_out;

  const int B = out_size;               // 4096 (O == 1)
  const int T = in_sizes[0] / B;        // 512  (I == 1)
  const int tiles = B / 16;             // 256 single-wave workgroups

  rnn_wmma_f32_kernel<<<dim3(tiles), dim3(32), 0, stream>>>(
      x, w_ih, w_hh, b_ih, b_hh, w_fc, b_fc, out, T);
}